// MoonshotLinear_24721831756590
// MI455X (gfx1250) — compile-verified
//
#include <hip/hip_runtime.h>

typedef float v2f __attribute__((ext_vector_type(2)));
typedef float v4f __attribute__((ext_vector_type(4)));
typedef float v8f __attribute__((ext_vector_type(8)));
typedef int   v4i_vs __attribute__((vector_size(16)));            // param type of async-LDS builtin
typedef __attribute__((address_space(3))) v4i_vs* lds_v4i_p;      // LDS-side param type

#define M_TOTAL 8192
#define N_TOTAL 4096
#define K_TOTAL 4096
#define TILE_M  128
#define TILE_N  128
#define TILE_K  32
#define KITERS  (K_TOTAL / TILE_K)
#define APITCH  36   // floats per A row in LDS (16B-aligned rows, conflict-free frag reads)
#define BPITCH  36   // floats per B row in LDS

// CDNA5 async global->LDS path (ASYNCcnt-tracked DMA), guarded for toolchain support.
#if __has_builtin(__builtin_amdgcn_global_load_async_to_lds_b128) && \
    __has_builtin(__builtin_amdgcn_s_wait_asynccnt)
#define USE_ASYNC_LDS 1
#else
#define USE_ASYNC_LDS 0
#endif

// C[8192,4096] = X[8192,4096] * W^T + bias, W dequantized from PQ codebook.
// 256 threads = 8 wave32 per workgroup; 128x128 C tile; waves 2(M) x 4(N),
// each wave owns 64x32 = 4x2 WMMA 16x16 tiles. Double-buffered LDS pipeline:
//   dequant B(it) -> wait async A(it) -> barrier -> issue async A(it+1) -> WMMA(it)
__global__ __launch_bounds__(256)
void pq_linear_wmma_f32(const float* __restrict__ x,
                        const int*   __restrict__ indices,
                        const float* __restrict__ codebook,
                        const float* __restrict__ bias,
                        float*       __restrict__ out)
{
    __shared__ float As[2][TILE_M * APITCH];   // A tiles, [m][k] row-major
    __shared__ float Bs[2][TILE_N * BPITCH];   // W tiles, [n][k] row-major (dequantized)
    __shared__ float CB[256 * 8];              // codebook resident in LDS

    const int t    = threadIdx.x;
    const int lane = t & 31;
    const int wid  = t >> 5;
    const int half = lane >> 4;   // 0: lanes 0-15, 1: lanes 16-31
    const int lr   = lane & 15;
    const int wm   = wid >> 2;    // 0..1  (64 rows each)
    const int wn   = wid & 3;     // 0..3  (32 cols each)

    const int mBase = blockIdx.y * TILE_M;
    const int nBase = blockIdx.x * TILE_N;

    // ---- preload codebook (8 KB) into LDS ----
    {
        const v4f* cbg = (const v4f*)codebook;
        v4f*       cbl = (v4f*)CB;
        cbl[t * 2 + 0] = cbg[t * 2 + 0];
        cbl[t * 2 + 1] = cbg[t * 2 + 1];
    }

    // Per-thread A staging geometry: 4 rows (rbase + p*32), one 16B chunk each.
    const int colv  = (t & 7) * 4;   // 0..28
    const int rbase = t >> 3;        // 0..31

    // ---- prologue: stage A(0) into As[0] ----
#if USE_ASYNC_LDS
#pragma unroll
    for (int p = 0; p < 4; ++p) {
        const int row = rbase + p * 32;
        __builtin_amdgcn_global_load_async_to_lds_b128(
            (v4i_vs*)(x + (size_t)(mBase + row) * K_TOTAL + colv),
            (lds_v4i_p)(&As[0][row * APITCH + colv]),
            0, 0);
    }
#else
    v4f areg[4];
#pragma unroll
    for (int p = 0; p < 4; ++p) {
        const int row = rbase + p * 32;
        areg[p] = *(const v4f*)(x + (size_t)(mBase + row) * K_TOTAL + colv);
    }
#endif

    // ---- prologue: prefetch PQ codes for it=0 ----
    int idxreg[2];
#pragma unroll
    for (int p = 0; p < 2; ++p) {
        const int s = t + p * 256;                 // 0..511
        idxreg[p] = indices[(nBase + (s >> 2)) * (K_TOTAL / 8) + (s & 3)];
    }

    v8f acc[4][2];
    const v8f zero = {0.f, 0.f, 0.f, 0.f, 0.f, 0.f, 0.f, 0.f};
#pragma unroll
    for (int mt = 0; mt < 4; ++mt)
#pragma unroll
        for (int nt = 0; nt < 2; ++nt)
            acc[mt][nt] = zero;

    __syncthreads();  // codebook visible to all waves

    for (int it = 0; it < KITERS; ++it) {
        const int buf = it & 1;
        const int itn = (it + 1 < KITERS) ? (it + 1) : it;  // clamped next iter
        const int kn  = itn * TILE_K;

        // (A) dequantize B(it) into Bs[buf] from prefetched codes; prefetch codes(it+1)
#pragma unroll
        for (int p = 0; p < 2; ++p) {
            const int s  = t + p * 256;        // 0..511
            const int nl = s >> 2;             // 0..127
            const int j  = s & 3;              // which 8-wide subvec in k-tile
            const float* c = &CB[idxreg[p] * 8];
            float* dst = &Bs[buf][nl * BPITCH + j * 8];
#pragma unroll
            for (int q = 0; q < 4; ++q) {
                v2f v;
                v.x = c[2 * q + 0];
                v.y = c[2 * q + 1];
                *(v2f*)(dst + 2 * q) = v;      // 8B-aligned ds_store_b64
            }
            idxreg[p] = indices[(nBase + nl) * (K_TOTAL / 8) + (kn >> 3) + j];
        }

        // (B) make A(it) resident in As[buf]
#if USE_ASYNC_LDS
        __builtin_amdgcn_s_wait_asynccnt(0);   // async DMA for A(it) landed
#else
#pragma unroll
        for (int p = 0; p < 4; ++p) {
            const int row = rbase + p * 32;
            *(v4f*)(&As[buf][row * APITCH + colv]) = areg[p];
        }
#endif

        // (C) all waves done with buffers {buf^1 reads from it-1, buf writes from this phase}
        __syncthreads();

        // (D) stage A(it+1) into As[buf^1] while WMMAs run
#if USE_ASYNC_LDS
#pragma unroll
        for (int p = 0; p < 4; ++p) {
            const int row = rbase + p * 32;
            __builtin_amdgcn_global_load_async_to_lds_b128(
                (v4i_vs*)(x + (size_t)(mBase + row) * K_TOTAL + kn + colv),
                (lds_v4i_p)(&As[buf ^ 1][row * APITCH + colv]),
                0, 0);
        }
#else
#pragma unroll
        for (int p = 0; p < 4; ++p) {
            const int row = rbase + p * 32;
            areg[p] = *(const v4f*)(x + (size_t)(mBase + row) * K_TOTAL + kn + colv);
        }
#endif

        // (E) MMA phase: 8 k-substeps of 4, 8 WMMAs each
#pragma unroll
        for (int kk = 0; kk < TILE_K; kk += 4) {
            // A fragment: 16x4, VGPR0 = K{0,2} per half, VGPR1 = K{1,3}
            v2f afrag[4];
#pragma unroll
            for (int mt = 0; mt < 4; ++mt) {
                const int row = wm * 64 + mt * 16 + lr;
                afrag[mt] = *(const v2f*)(&As[buf][row * APITCH + kk + 2 * half]);
            }
            // B fragment: 4x16, mirrored layout from [n][k] tile
            v2f bfrag[2];
#pragma unroll
            for (int nt = 0; nt < 2; ++nt) {
                const int col = wn * 32 + nt * 16 + lr;
                bfrag[nt] = *(const v2f*)(&Bs[buf][col * BPITCH + kk + 2 * half]);
            }
#pragma unroll
            for (int mt = 0; mt < 4; ++mt)
#pragma unroll
                for (int nt = 0; nt < 2; ++nt)
                    acc[mt][nt] = __builtin_amdgcn_wmma_f32_16x16x4_f32(
                        /*neg_a=*/false, afrag[mt],
                        /*neg_b=*/false, bfrag[nt],
                        /*c_mod=*/(short)0, acc[mt][nt],
                        /*reuse_a=*/false, /*reuse_b=*/false);
        }
    }

    // ---- epilogue: C/D layout VGPR i -> M = i (lanes 0-15) / M = 8+i (lanes 16-31) ----
#pragma unroll
    for (int nt = 0; nt < 2; ++nt) {
        const int n  = nBase + wn * 32 + nt * 16 + lr;
        const float bv = bias[n];
#pragma unroll
        for (int mt = 0; mt < 4; ++mt) {
#pragma unroll
            for (int i = 0; i < 8; ++i) {
                const int row = mBase + wm * 64 + mt * 16 + half * 8 + i;
                out[(size_t)row * N_TOTAL + n] = acc[mt][nt][i] + bv;
            }
        }
    }
}

extern "C" void kernel_launch(void* const* d_in, const int* in_sizes, int n_in,
                              void* d_out, int out_size, void* d_ws, size_t ws_size,
                              hipStream_t stream) {
    const float* x        = (const float*)d_in[0];  // [4,2048,4096] -> [8192,4096]
    const int*   indices  = (const int*)d_in[1];    // [2097152]
    const float* codebook = (const float*)d_in[2];  // [256,8]
    const float* bias     = (const float*)d_in[3];  // [4096]
    float*       out      = (float*)d_out;          // [8192,4096]

    dim3 grid(N_TOTAL / TILE_N, M_TOTAL / TILE_M);  // (32, 64)
    dim3 block(256);
    pq_linear_wmma_f32<<<grid, block, 0, stream>>>(x, indices, codebook, bias, out);
}